// ConvAttn_17901423690449
// MI455X (gfx1250) — compile-verified
//
#include <hip/hip_runtime.h>

// ---------------------------------------------------------------------------
// ConvAttn on MI455X (gfx1250): fused flash-attention, f32 WMMA + TDM staging.
//   Kernel 1: qkv projection -> workspace q,k,v [b][32][4096]
//   Kernel 2: block of 8 waves; K/V tiles (32x16 f32) staged into LDS by the
//             Tensor Data Mover (double-buffered, s_wait_tensorcnt), each wave
//             runs flash attention for its own 16-query tile with
//             V_WMMA_F32_16X16X4_F32; softmax row-sum accumulated via a
//             P x ones WMMA accumulator; fused Wp projection + residual.
// ---------------------------------------------------------------------------

typedef float v2f __attribute__((ext_vector_type(2)));
typedef float v8f __attribute__((ext_vector_type(8)));
typedef unsigned int u32x4 __attribute__((ext_vector_type(4)));
typedef int i32x4 __attribute__((ext_vector_type(4)));
typedef int i32x8 __attribute__((ext_vector_type(8)));

#define NTOK 4096
#define CIN  64
#define HID  32
#define NTILE (NTOK / 16)

__device__ __forceinline__ v8f wmma_f32(v2f a, v2f b, v8f c) {
  // (neg_a, A, neg_b, B, c_mod, C, reuse_a, reuse_b)
  return __builtin_amdgcn_wmma_f32_16x16x4_f32(false, a, false, b, (short)0, c,
                                               false, false);
}

// TDM: load one 32(row) x 16(col) f32 tile of a [32][4096] row-major tensor
// into LDS (contiguous [32][16]).  D# per CDNA5 ISA ch.8 (2D, groups 2/3 zero).
__device__ __forceinline__ void tdm_load_tile(unsigned int lds_off,
                                              const float* tile_gaddr) {
  unsigned long long ga = (unsigned long long)(uintptr_t)tile_gaddr;
  u32x4 g0;
  g0[0] = 1u;                                       // count=1, user mode
  g0[1] = lds_off;                                  // lds_addr (bytes)
  g0[2] = (unsigned int)(ga & 0xFFFFFFFFu);         // global_addr[31:0]
  g0[3] = (unsigned int)((ga >> 32) & 0x01FFFFFFu)  // global_addr[56:32]
          | (2u << 30);                             // type=2 ("image")
  i32x8 g1;
  g1[0] = (int)(2u << 16);          // wg_mask=0, data_size=2 (4 bytes)
  g1[1] = (int)(4096u << 16);       // tensor_dim0 = 4096   (bits 79:48)
  g1[2] = (int)(32u << 16);         // tensor_dim1 = 32     (bits 111:80)
  g1[3] = (int)(16u << 16);         // tile_dim0   = 16     (bits 127:112)
  g1[4] = 32;                       // tile_dim1   = 32     (bits 143:128)
  g1[5] = 4096;                     // tensor_dim0_stride   (bits 207:160)
  g1[6] = 0;                        // stride hi / dim1_stride lo
  g1[7] = 0;
  i32x4 z4 = {0, 0, 0, 0};
  i32x8 z8 = {0, 0, 0, 0, 0, 0, 0, 0};
  // 6-arg form (clang-23 / therock-10.0): (g0, g1, g2, g3, g4, cpol)
  __builtin_amdgcn_tensor_load_to_lds(g0, g1, z4, z4, z8, 0);
}

// ---------------------------------------------------------------------------
// Kernel 1: q/k/v = W @ x + bias.  grid (16, 96, 4), block 256.
// ---------------------------------------------------------------------------
__global__ __launch_bounds__(256)
void qkv_proj_kernel(const float* __restrict__ x,
                     const float* __restrict__ wq, const float* __restrict__ bq,
                     const float* __restrict__ wk, const float* __restrict__ bk,
                     const float* __restrict__ wv, const float* __restrict__ bv,
                     float* __restrict__ qo, float* __restrict__ ko,
                     float* __restrict__ vo)
{
  const int n = blockIdx.x * 256 + threadIdx.x;
  const int o = blockIdx.y;
  const int b = blockIdx.z;

  const float* w;  const float* bias;  float* outp;  int oo;
  if (o < 32)       { w = wq; bias = bq; outp = qo; oo = o; }
  else if (o < 64)  { w = wk; bias = bk; outp = ko; oo = o - 32; }
  else              { w = wv; bias = bv; outp = vo; oo = o - 64; }

  const float* xb = x + (size_t)b * CIN * NTOK + n;
  float acc = bias[oo];
#pragma unroll
  for (int c = 0; c < CIN; ++c)
    acc = fmaf(w[oo * CIN + c], xb[(size_t)c * NTOK], acc);

  outp[((size_t)b * HID + oo) * NTOK + n] = acc;
}

// ---------------------------------------------------------------------------
// Kernel 2: fused flash attention + projection + residual.
// grid (32, 4), block 256 = 8 waves; each wave owns one 16-row query tile;
// the block shares TDM-staged K/V tiles.
// ---------------------------------------------------------------------------
__global__ __launch_bounds__(256, 2)
void attn_proj_kernel(const float* __restrict__ qg, const float* __restrict__ kg,
                      const float* __restrict__ vg, const float* __restrict__ xg,
                      const float* __restrict__ wp, const float* __restrict__ bp,
                      float* __restrict__ outg)
{
  __shared__ float kvbuf[2][2][HID * 16];   // [buf][K/V][32 rows x 16 cols]
  __shared__ float pstage[8][512];          // per-wave P / z staging

  const int lane  = threadIdx.x & 31;
  const int wv_id = threadIdx.x >> 5;
  const int ln    = lane & 15;              // row (A) / col (B,D) index
  const int hi    = lane >> 4;              // lane-half: K-pair / row+8 select
  const int kb2   = hi * 2;
  const int bz    = blockIdx.y;
  const int n0    = (blockIdx.x * 8 + wv_id) * 16;

  const float* qb  = qg + (size_t)bz * HID * NTOK;
  const float* kbp = kg + (size_t)bz * HID * NTOK;
  const float* vbp = vg + (size_t)bz * HID * NTOK;
  float* pbuf = pstage[wv_id];

  // ---- Q tile in A layout: 8 K-chunks of 4 over hid ----------------------
  v2f qa[8];
#pragma unroll
  for (int kc = 0; kc < 8; ++kc) {
    const int h = kc * 4 + kb2;
    qa[kc].x = qb[(size_t)h * NTOK + n0 + ln];
    qa[kc].y = qb[(size_t)(h + 1) * NTOK + n0 + ln];
  }

  v8f o0 = {0,0,0,0,0,0,0,0};               // z cols h=0..15
  v8f o1 = {0,0,0,0,0,0,0,0};               // z cols h=16..31
  v8f ol = {0,0,0,0,0,0,0,0};               // softmax denominator (P x ones)
  const v2f ones = {1.0f, 1.0f};
  float mrow[8];
#pragma unroll
  for (int r = 0; r < 8; ++r) mrow[r] = -3.0e38f;

  // ---- prime TDM double buffer -------------------------------------------
  if (wv_id == 0) {
    tdm_load_tile((unsigned int)(uintptr_t)&kvbuf[0][0][0], kbp);
    tdm_load_tile((unsigned int)(uintptr_t)&kvbuf[0][1][0], vbp);
  }

  // ---- flash loop over 256 key tiles -------------------------------------
  for (int t = 0; t < NTILE; ++t) {
    const int buf = t & 1;
    if (wv_id == 0) {
      if (t + 1 < NTILE) {
        // prefetch next tile into the other buffer (prev iter's end-barrier
        // guarantees no wave still reads it), then wait for current tile:
        // TDM ops are in-order per wave, so tensorcnt<=2 => tile t landed.
        tdm_load_tile((unsigned int)(uintptr_t)&kvbuf[buf ^ 1][0][0],
                      kbp + (t + 1) * 16);
        tdm_load_tile((unsigned int)(uintptr_t)&kvbuf[buf ^ 1][1][0],
                      vbp + (t + 1) * 16);
        __builtin_amdgcn_s_wait_tensorcnt(2);
      } else {
        __builtin_amdgcn_s_wait_tensorcnt(0);
      }
    }
    __syncthreads();                         // current tile visible to block
    const float* ldsK = kvbuf[buf][0];
    const float* ldsV = kvbuf[buf][1];

    // S = Q(16x32) x K(32x16): 8 chained f32 WMMAs, B fed from LDS
    v8f s = {0,0,0,0,0,0,0,0};
#pragma unroll
    for (int kc = 0; kc < 8; ++kc) {
      const int h = kc * 4 + kb2;
      v2f bb;
      bb.x = ldsK[h * 16 + ln];
      bb.y = ldsK[(h + 1) * 16 + ln];
      s = wmma_f32(qa[kc], bb, s);
    }

    // online softmax: row max via xor-shuffles within each 16-lane half;
    // row-sum is NOT shuffled -- it is accumulated by the P x ones WMMA.
#pragma unroll
    for (int r = 0; r < 8; ++r) {
      float sv = s[r];
      float mx = sv;
      mx = fmaxf(mx, __shfl_xor(mx, 1, 32));
      mx = fmaxf(mx, __shfl_xor(mx, 2, 32));
      mx = fmaxf(mx, __shfl_xor(mx, 4, 32));
      mx = fmaxf(mx, __shfl_xor(mx, 8, 32));
      const float mnew = fmaxf(mrow[r], mx);
      const float scl  = __expf(mrow[r] - mnew);
      const float p    = __expf(sv - mnew);
      mrow[r] = mnew;
      o0[r] *= scl;
      o1[r] *= scl;
      ol[r] *= scl;
      // stage P (C/D layout -> row-major [16][16]) for A-layout re-read
      pbuf[(hi * 8 + r) * 16 + ln] = p;
    }
    __builtin_amdgcn_wave_barrier();
    asm volatile("s_wait_dscnt 0" ::: "memory");

    // O += P(16x16) x V^T(16x32); L += P x ones
#pragma unroll
    for (int j = 0; j < 4; ++j) {
      const int mk = j * 4 + kb2;
      v2f pa  = *(const v2f*)(pbuf + ln * 16 + mk);
      v2f vb0 = *(const v2f*)(ldsV + ln * 16 + mk);
      v2f vb1 = *(const v2f*)(ldsV + (16 + ln) * 16 + mk);
      o0 = wmma_f32(pa, vb0, o0);
      o1 = wmma_f32(pa, vb1, o1);
      ol = wmma_f32(pa, ones, ol);
    }
    __syncthreads();                         // all waves done with this buffer
  }

  // ---- finalize z = O / L, stage to LDS as z[n][h] ------------------------
#pragma unroll
  for (int r = 0; r < 8; ++r) {
    const float inv = 1.0f / ol[r];
    const int nr = hi * 8 + r;
    pbuf[nr * 32 + ln]      = o0[r] * inv;
    pbuf[nr * 32 + 16 + ln] = o1[r] * inv;
  }
  __builtin_amdgcn_wave_barrier();
  asm volatile("s_wait_dscnt 0" ::: "memory");

  // ---- out = x + Wp(64x32) x z^T(32x16) + bp ------------------------------
#pragma unroll
  for (int ct = 0; ct < 4; ++ct) {
    v8f acc = {0,0,0,0,0,0,0,0};
#pragma unroll
    for (int kc = 0; kc < 8; ++kc) {
      const int h = kc * 4 + kb2;
      v2f wa = *(const v2f*)(wp + (ct * 16 + ln) * HID + h);   // Wp[c][h]
      v2f zb = *(const v2f*)(pbuf + ln * 32 + h);              // z[n][h]
      acc = wmma_f32(wa, zb, acc);
    }
#pragma unroll
    for (int r = 0; r < 8; ++r) {
      const int c = ct * 16 + hi * 8 + r;
      const size_t idx = ((size_t)bz * CIN + c) * NTOK + n0 + ln;
      outg[idx] = xg[idx] + bp[c] + acc[r];
    }
  }
}

// ---------------------------------------------------------------------------
extern "C" void kernel_launch(void* const* d_in, const int* in_sizes, int n_in,
                              void* d_out, int out_size, void* d_ws, size_t ws_size,
                              hipStream_t stream) {
  const float* x  = (const float*)d_in[0];
  const float* wq = (const float*)d_in[1];
  const float* bq = (const float*)d_in[2];
  const float* wk = (const float*)d_in[3];
  const float* bk = (const float*)d_in[4];
  const float* wv = (const float*)d_in[5];
  const float* bv = (const float*)d_in[6];
  const float* wp = (const float*)d_in[7];
  const float* bp = (const float*)d_in[8];
  float* out = (float*)d_out;

  const size_t per = (size_t)4 * HID * NTOK;   // 524288 floats = 2 MB each
  float* qws = (float*)d_ws;
  float* kws = qws + per;
  float* vws = kws + per;

  qkv_proj_kernel<<<dim3(NTOK / 256, 96, 4), 256, 0, stream>>>(
      x, wq, bq, wk, bk, wv, bv, qws, kws, vws);

  attn_proj_kernel<<<dim3(NTOK / 16 / 8, 4), 256, 0, stream>>>(
      qws, kws, vws, x, wp, bp, out);
}